// SimpleKP_70042326663876
// MI455X (gfx1250) — compile-verified
//
#include <hip/hip_runtime.h>
#include <cstdint>
#include <cstddef>

#define AS1 __attribute__((address_space(1)))
#define AS3 __attribute__((address_space(3)))

typedef float        f32x4 __attribute__((ext_vector_type(4)));
typedef float        f32x2 __attribute__((ext_vector_type(2)));
typedef unsigned int u32x4 __attribute__((ext_vector_type(4)));
typedef int          i32x4 __attribute__((ext_vector_type(4)));
typedef int          i32x8 __attribute__((ext_vector_type(8)));

#define KS     5
#define W      512
#define H      512
#define OW     508
#define NBATCH 4
#define PLANE  (W * H)
#define EPSV   1e-4f

#if __has_builtin(__builtin_amdgcn_tensor_load_to_lds) && __has_builtin(__builtin_amdgcn_s_wait_tensorcnt)
#define HAVE_TDM 1
#else
#define HAVE_TDM 0
#endif

#if __has_builtin(__builtin_amdgcn_global_load_async_to_lds_b128)
#define HAVE_ASYNC 1
#else
#define HAVE_ASYNC 0
#endif

// ---------------------------------------------------------------------------
// Main pass (defined FIRST so the disasm snippet shows it): one block per
// (output row y, b*3+i).  128 threads, 4 pixels/thread, pixel group
// x = 4t-2 .. 4t+1 so every `kernels` b128 load (col x+2 = 4t..4t+3) is
// 16-byte aligned.  Gray rows y..y+4 are staged into LDS once per block by a
// single Tensor Data Mover 2D-tile descriptor (512 x 5 f32, stride 512).
// ---------------------------------------------------------------------------
__global__ __launch_bounds__(128) void kpn_kernel(const float* __restrict__ kernels,
                                                  const float* __restrict__ gray,
                                                  float* __restrict__ out) {
  __shared__ float sgray[KS * W];                     // 5 rows x 512 cols = 10 KB

  const int y  = blockIdx.x;                          // 0..507
  const int bi = blockIdx.y;                          // 0..11
  const int b  = bi / 3;
  const int i  = bi % 3;
  const int t  = threadIdx.x;                         // 0..127

  const float* gbase = gray + (size_t)b * PLANE + (size_t)y * W;  // rows y..y+4

#if HAVE_TDM
  // --- Tensor Data Mover: one 2D tile load (512 x 5 f32, stride 512) per block.
  if (t < 32) {                                       // wave 0 only issues the DMA
    unsigned lds_off = (unsigned)(uintptr_t)sgray;    // low 32 bits = LDS byte offset
    uint64_t ga      = (uint64_t)(uintptr_t)gbase;    // global byte address of tile
    u32x4 g0 = { 1u,                                  // count=1 valid descriptor
                 lds_off,                              // lds_addr
                 (unsigned)ga,                         // global_addr[31:0]
                 ((unsigned)(ga >> 32) & 0x01FFFFFFu) | 0x80000000u }; // addr[56:32] | type=2
    i32x8 g1 = { (int)(2u << 16),                     // data_size = 4 bytes
                 (int)(512u << 16),                   // tensor_dim0 lo16 (=512)
                 (int)(512u << 16),                   // dim0 hi16=0 | tensor_dim1 lo16 (=512)
                 (int)(512u << 16),                   // dim1 hi16=0 | tile_dim0 = 512
                 5,                                   // tile_dim1 = 5, tile_dim2 = 0
                 512,                                 // tensor_dim0_stride lo32 = 512
                 0, 0 };
    i32x4 z4 = { 0, 0, 0, 0 };
#if defined(__clang_major__) && (__clang_major__ >= 23)
    i32x8 z8 = { 0, 0, 0, 0, 0, 0, 0, 0 };
    __builtin_amdgcn_tensor_load_to_lds(g0, g1, z4, z4, z8, 0);
#else
    __builtin_amdgcn_tensor_load_to_lds(g0, g1, z4, z4, 0);
#endif
    __builtin_amdgcn_s_wait_tensorcnt(0);
  }
  __syncthreads();
#elif HAVE_ASYNC
  // --- Async global->LDS copy path (per-lane b128), ASYNCcnt tracked.
  {
    #pragma unroll
    for (int r = 0; r < KS; ++r) {
      const AS1 void* src = (const AS1 void*)(uintptr_t)(gbase + r * W + 4 * t);
      AS3 void* dst = (AS3 void*)(uintptr_t)((unsigned)(uintptr_t)(sgray + r * W + 4 * t));
      __builtin_amdgcn_global_load_async_to_lds_b128(src, dst, 0, 0);
    }
#if __has_builtin(__builtin_amdgcn_s_wait_asynccnt)
    __builtin_amdgcn_s_wait_asynccnt(0);
#elif defined(__AMDGCN__)
    asm volatile("s_wait_asynccnt 0" ::: "memory");
#endif
  }
  __syncthreads();
#else
  // --- Plain cooperative staging fallback.
  {
    #pragma unroll
    for (int r = 0; r < KS; ++r)
      *(f32x4*)(sgray + r * W + 4 * t) = *(const f32x4*)(gbase + r * W + 4 * t);
  }
  __syncthreads();
#endif

  // kernels plane base: channel (i*25 + j), row y+2, col 4t (16B aligned, in-bounds:
  // 4t+3 <= 511, y+2 <= 509).
  const float* kbase = kernels
      + ((size_t)(b * 75 + i * 25) * H + (size_t)(y + 2)) * W + 4 * t;

  float dot0 = 0.f, dot1 = 0.f, dot2 = 0.f, dot3 = 0.f;
  float nrm0 = 0.f, nrm1 = 0.f, nrm2 = 0.f, nrm3 = 0.f;

  // gray window for this thread: cols 4t-4 .. 4t+7 (12 floats) per row, loaded as
  // three aligned ds b128.  Clamped chunks only feed the out-of-range pixels,
  // which are never stored.
  int c0 = 4 * t - 4; if (c0 < 0)   c0 = 0;
  int c2 = 4 * t + 4; if (c2 > 508) c2 = 508;
  const int c1 = 4 * t;

  #pragma unroll
  for (int dy = 0; dy < KS; ++dy) {
    const float* srow = sgray + dy * W;
    f32x4 ga = *(const f32x4*)(srow + c0);
    f32x4 gb = *(const f32x4*)(srow + c1);
    f32x4 gc = *(const f32x4*)(srow + c2);
    float g[12] = { ga.x, ga.y, ga.z, ga.w,
                    gb.x, gb.y, gb.z, gb.w,
                    gc.x, gc.y, gc.z, gc.w };   // g[k] == gray[y+dy][4t-4+k]
    #pragma unroll
    for (int dx = 0; dx < KS; ++dx) {
      const int j = dy * KS + dx;
      f32x4 kv = __builtin_nontemporal_load((const f32x4*)(kbase + (size_t)j * PLANE));
      // pixel p (x = 4t-2+p) uses kv[p] and gray col x+dx -> g[p+dx+2]
      nrm0 += __builtin_fabsf(kv.x);  dot0 = __builtin_fmaf(kv.x, g[dx + 2], dot0);
      nrm1 += __builtin_fabsf(kv.y);  dot1 = __builtin_fmaf(kv.y, g[dx + 3], dot1);
      nrm2 += __builtin_fabsf(kv.z);  dot2 = __builtin_fmaf(kv.z, g[dx + 4], dot2);
      nrm3 += __builtin_fabsf(kv.w);  dot3 = __builtin_fmaf(kv.w, g[dx + 5], dot3);
    }
  }

#if __has_builtin(__builtin_amdgcn_rcpf)
  float r0 = dot0 * __builtin_amdgcn_rcpf(nrm0 + EPSV);
  float r1 = dot1 * __builtin_amdgcn_rcpf(nrm1 + EPSV);
  float r2 = dot2 * __builtin_amdgcn_rcpf(nrm2 + EPSV);
  float r3 = dot3 * __builtin_amdgcn_rcpf(nrm3 + EPSV);
#else
  float r0 = dot0 / (nrm0 + EPSV);
  float r1 = dot1 / (nrm1 + EPSV);
  float r2 = dot2 / (nrm2 + EPSV);
  float r3 = dot3 / (nrm3 + EPSV);
#endif

  float* orow = out + ((size_t)(b * 3 + i) * OW + y) * OW;
  if (t > 0) {          // pixels x = 4t-2, 4t-1  (t=0 would be x=-2,-1)
    f32x2 v = { r0, r1 };
    __builtin_nontemporal_store(v, (f32x2*)(orow + 4 * t - 2));
  }
  if (t < 127) {        // pixels x = 4t, 4t+1    (t=127 would be x=508,509)
    f32x2 v = { r2, r3 };
    __builtin_nontemporal_store(v, (f32x2*)(orow + 4 * t));
  }
}

// ---------------------------------------------------------------------------
// Pass 1: gray[b,y,x] = sum_c mosaic[b,c,y,x].  Pure streaming, float4, NT in.
// Grid: 1024 x 256 threads covers 4*512*512/4 = 262144 float4s exactly.
// ---------------------------------------------------------------------------
__global__ __launch_bounds__(256) void gray_sum_kernel(const float* __restrict__ mosaic,
                                                       float* __restrict__ gray) {
  int idx = blockIdx.x * blockDim.x + threadIdx.x;   // 0 .. 262143
  int b   = idx >> 16;                                // PLANE/4 == 65536
  int p   = idx & 0xFFFF;
  const f32x4* m = (const f32x4*)mosaic + (size_t)(b * 3) * (PLANE / 4) + p;
  f32x4 g = __builtin_nontemporal_load(m)
          + __builtin_nontemporal_load(m + (PLANE / 4))
          + __builtin_nontemporal_load(m + 2 * (PLANE / 4));
  ((f32x4*)gray)[idx] = g;   // keep resident (RT) — pass 2 re-reads it 5x from L2
}

// ---------------------------------------------------------------------------
extern "C" void kernel_launch(void* const* d_in, const int* in_sizes, int n_in,
                              void* d_out, int out_size, void* d_ws, size_t ws_size,
                              hipStream_t stream) {
  const float* mosaic  = (const float*)d_in[0];   // (4,3,512,512) f32
  const float* kernels = (const float*)d_in[1];   // (4,75,512,512) f32
  float*       gray    = (float*)d_ws;            // 4*512*512 f32 = 4 MB scratch
  float*       out     = (float*)d_out;           // (4,3,508,508) f32

  gray_sum_kernel<<<dim3(1024), dim3(256), 0, stream>>>(mosaic, gray);
  kpn_kernel<<<dim3(OW, NBATCH * 3), dim3(128), 0, stream>>>(kernels, gray, out);
}